// AllTimes_39290360824569
// MI455X (gfx1250) — compile-verified
//
#include <hip/hip_runtime.h>
#include <cstdint>

#define NBANDS  5
#define BATCH   256
#define LEN     800
#define TTOT    (NBANDS * LEN)      // 4000
#define NOUT    (TTOT - NBANDS)     // 3995 = T - (NBANDS-1) - 1
#define SORTN   4096
#define THREADS 256
#define R       (SORTN / THREADS)   // 16 elements per thread

__device__ __forceinline__ void cx(float& a, float& b, bool up) {
    float lo = fminf(a, b);
    float hi = fmaxf(a, b);
    a = up ? lo : hi;
    b = up ? hi : lo;
}

// One bitonic stage (sub-passes j = K/2 .. 1) with three regimes:
//   j >= 256         : partner in same thread's registers (pure VALU)
//   j in {128,64,32} : cross-wave -> ONE fused LDS round-trip; each thread
//                      owns whole 8-element groups (index bits 5-7) and runs
//                      all applicable layers in registers
//   j <= 16          : partner in same wave32 -> __shfl_xor, no barrier
template<int K>
__device__ __forceinline__ void bitonic_stage(float (&e)[R], float* s, int tid)
{
    // (1) register-local passes, j >= 256
    #pragma unroll
    for (int j = K / 2; j >= THREADS; j >>= 1) {
        const int jm = j / THREADS;                 // compile-time constant
        #pragma unroll
        for (int m = 0; m < R; ++m) {
            if ((m & jm) == 0) {
                bool up = (((m * THREADS + tid) & K) == 0);
                cx(e[m], e[m + jm], up);
            }
        }
    }

    // (2) fused cross-wave passes j = {128,64,32} via LDS
    if constexpr (K >= 64) {
        #pragma unroll
        for (int m = 0; m < R; ++m) s[m * THREADS + tid] = e[m];
        __syncthreads();
        #pragma unroll
        for (int gg = 0; gg < 2; ++gg) {
            const int g    = tid + gg * THREADS;         // 512 groups of 8
            const int base = ((g >> 5) << 8) | (g & 31); // bits 5-7 free in group
            float r[8];
            #pragma unroll
            for (int bq = 0; bq < 8; ++bq) r[bq] = s[base + (bq << 5)];
            if constexpr (K >= 256) {                    // layer j = 128 (bit 7)
                #pragma unroll
                for (int bq = 0; bq < 4; ++bq) {
                    bool up = (((base + (bq << 5)) & K) == 0);
                    cx(r[bq], r[bq + 4], up);
                }
            }
            if constexpr (K >= 128) {                    // layer j = 64 (bit 6)
                #pragma unroll
                for (int bq = 0; bq < 8; ++bq) {
                    if ((bq & 2) == 0) {
                        bool up = (((base + (bq << 5)) & K) == 0);
                        cx(r[bq], r[bq + 2], up);
                    }
                }
            }
            #pragma unroll
            for (int bq = 0; bq < 8; bq += 2) {          // layer j = 32 (bit 5)
                bool up = (((base + (bq << 5)) & K) == 0);
                cx(r[bq], r[bq + 1], up);
            }
            #pragma unroll
            for (int bq = 0; bq < 8; ++bq) s[base + (bq << 5)] = r[bq];
        }
        __syncthreads();
        #pragma unroll
        for (int m = 0; m < R; ++m) e[m] = s[m * THREADS + tid];
    }

    // (3) intra-wave shuffle passes, j <= 16
    constexpr int JLOW = (K / 2 < 16) ? K / 2 : 16;
    #pragma unroll
    for (int j = JLOW; j >= 1; j >>= 1) {
        #pragma unroll
        for (int m = 0; m < R; ++m) {
            bool up = (((m * THREADS + tid) & K) == 0);
            float p = __shfl_xor(e[m], j, 32);
            bool keep_min = (((tid & j) == 0) == up);
            e[m] = keep_min ? fminf(e[m], p) : fmaxf(e[m], p);
        }
    }
}

__global__ __launch_bounds__(THREADS)
void AllTimes_sort_diff_kernel(const float* __restrict__ in,
                               const int*   __restrict__ ntotal,
                               float*       __restrict__ out)
{
    __shared__ float s[SORTN];
    const int b   = blockIdx.x;
    const int tid = threadIdx.x;

    // ---- Async global -> LDS load of 5 bands x 800 floats (B128 chunks) ----
    for (int band = 0; band < NBANDS; ++band) {
        const float* base = in + (size_t)band * (BATCH * LEN) + (size_t)b * LEN;
        if (tid < LEN / 4) {                        // 200 float4 chunks per band
            int      goff = tid * 16;               // byte offset from base
            unsigned lds  = (unsigned)(uintptr_t)(&s[band * LEN + tid * 4]);
            asm volatile("global_load_async_to_lds_b128 %0, %1, %2"
                         :: "v"(lds), "v"(goff), "s"(base)
                         : "memory");
        }
    }
    // Pad tail to power-of-two with +inf so padding sorts to the end.
    if (tid < SORTN - TTOT) s[TTOT + tid] = __builtin_inff();

    asm volatile("s_wait_asynccnt 0" ::: "memory");
    __syncthreads();

    // ---- Register-resident hybrid bitonic sort (ascending) ----
    float e[R];
    #pragma unroll
    for (int m = 0; m < R; ++m) e[m] = s[m * THREADS + tid];

    bitonic_stage<2>(e, s, tid);
    bitonic_stage<4>(e, s, tid);
    bitonic_stage<8>(e, s, tid);
    bitonic_stage<16>(e, s, tid);
    bitonic_stage<32>(e, s, tid);
    bitonic_stage<64>(e, s, tid);
    bitonic_stage<128>(e, s, tid);
    bitonic_stage<256>(e, s, tid);
    bitonic_stage<512>(e, s, tid);
    bitonic_stage<1024>(e, s, tid);
    bitonic_stage<2048>(e, s, tid);
    bitonic_stage<4096>(e, s, tid);

    // Publish fully sorted array for the neighbor-difference step.
    #pragma unroll
    for (int m = 0; m < R; ++m) s[m * THREADS + tid] = e[m];
    __syncthreads();

    // ---- diffs of sorted[4:] (drop 4 smallest), mask by N_total, store ----
    const int nt = ntotal[b];
    float* ob = out + (size_t)b * NOUT;
    for (int i = tid; i < NOUT; i += THREADS) {
        float d = s[i + NBANDS] - s[i + NBANDS - 1];   // st[i+1] - st[i]
        float v = (i < nt) ? d : 0.0f;
        __builtin_nontemporal_store(v, &ob[i]);        // 4 MB streaming write
    }
}

extern "C" void kernel_launch(void* const* d_in, const int* in_sizes, int n_in,
                              void* d_out, int out_size, void* d_ws, size_t ws_size,
                              hipStream_t stream)
{
    const float* in  = (const float*)d_in[0];   // (5, 256, 800) fp32
    const int*   nt  = (const int*)d_in[1];     // (256,) int32
    float*       out = (float*)d_out;           // (256, 3995, 1) fp32

    AllTimes_sort_diff_kernel<<<BATCH, THREADS, 0, stream>>>(in, nt, out);
}